// GCN_90013924590094
// MI455X (gfx1250) — compile-verified
//
#include <hip/hip_runtime.h>
#include <cstdint>

// Problem constants (match reference).
#define DIM     100
#define NNODES  50000
#define NEDGES  800000
#define NLAYERS 3
#define TILES_N 7          // ceil(100/16) column tiles

typedef __attribute__((ext_vector_type(2))) float v2f;
typedef __attribute__((ext_vector_type(8))) float v8f;
typedef __attribute__((ext_vector_type(4))) unsigned int u32x4;
typedef __attribute__((ext_vector_type(8))) int i32x8;
typedef __attribute__((ext_vector_type(4))) int i32x4;

// ---------------------------------------------------------------------------
// TDM: DMA the whole 100x100 f32 weight matrix (40 KB, contiguous) into LDS.
// Descriptor per CDNA5 ISA:
//   group0: count=1 | lds_addr | global_addr[56:0] | type=2
//   group1: data_size=4B, tensor_dim0=tile_dim0=10000, tensor_dim1=tile_dim1=1,
//           tensor_dim0_stride=10000, workgroup_mask=0 (not in a cluster)
// This toolchain exposes the 6-arg builtin:
//   (u32x4 g0, i32x8 g1, i32x4 g2, i32x4 g3, i32x8 extra, i32 cpol)
// Issued by ONE wave per block (TDM ignores EXEC; per-wave instruction).
// ---------------------------------------------------------------------------
__device__ inline void tdm_load_w_to_lds(const float* W, float* lds_w)
{
    const unsigned long long ga = (unsigned long long)(uintptr_t)W;
    const unsigned int lds_off = (unsigned int)(uintptr_t)lds_w; // low32 = LDS offset

    u32x4 g0;
    g0[0] = 1u;                                            // count=1, user mode
    g0[1] = lds_off;                                       // lds_addr (bytes)
    g0[2] = (unsigned int)(ga & 0xFFFFFFFFu);              // global_addr[31:0]
    g0[3] = (unsigned int)((ga >> 32) & 0x01FFFFFFu)       // global_addr[56:32]
          | (2u << 30);                                    // type=2 ("image")

    i32x8 g1;
    g1[0] = (int)(2u << 16);        // workgroup_mask=0, data_size=2 (4 bytes)
    g1[1] = (int)(10000u << 16);    // tensor_dim0[15:0] at bits 63:48
    g1[2] = (int)(1u << 16);        // tensor_dim0[31:16]=0 | tensor_dim1=1
    g1[3] = (int)(10000u << 16);    // tensor_dim1 hi=0 | tile_dim0=10000
    g1[4] = 1;                      // tile_dim1=1, tile_dim2=0
    g1[5] = 10000;                  // tensor_dim0_stride[31:0]
    g1[6] = 0;                      // stride hi, tensor_dim1_stride lo
    g1[7] = 0;                      // tensor_dim1_stride hi

    i32x4 z4 = {0, 0, 0, 0};        // groups 2/3 unused (<=2D tensor)
    i32x8 z8 = {0, 0, 0, 0, 0, 0, 0, 0};
    __builtin_amdgcn_tensor_load_to_lds(g0, g1, z4, z4, z8, 0);
    __builtin_amdgcn_s_wait_tensorcnt(0);                  // data landed in LDS
}

// ---------------------------------------------------------------------------
// GEMM with folded input mask/ReLU:
//   T[N,100] = f(X)[N,100] @ W[100,100]^T + b,
//   f(x_row) = (depth[row] <= thresh) ? (do_relu ? max(x,0) : x) : 0
// One wave32 per 16-row strip; 7 accumulators (one per 16-col tile); A loaded
// once per K-step from global, B fragments from LDS (ds_load_b64, no bank
// conflicts), 25*7 = 175 V_WMMA_F32_16X16X4_F32 per wave.
// EXEC stays all-1s through every WMMA: strip guard is wave-uniform, masking
// is v_cndmask, col clamp keeps LDS loads legal.
// ---------------------------------------------------------------------------
__global__ __launch_bounds__(256) void gcn_gemm_wmma(
    const float* __restrict__ X,
    const float* __restrict__ W,     // [100][100] row-major (this layer)
    const float* __restrict__ bias,  // [100]
    const int*   __restrict__ depth, // [N]
    int thresh, int do_relu,
    float* __restrict__ T)
{
    __shared__ float lds_w[DIM * DIM];                     // 40 KB

    if ((threadIdx.x >> 5) == 0)                           // wave 0 only
        tdm_load_w_to_lds(W, lds_w);
    __syncthreads();

    const int lane  = threadIdx.x & 31;
    const int strip = blockIdx.x * (blockDim.x >> 5) + (threadIdx.x >> 5);
    if (strip >= NNODES / 16) return;                      // wave-uniform exit

    const int l15  = lane & 15;
    const int koff = (lane >> 4) << 1;                     // 0 or 2
    const int row  = strip * 16 + l15;                     // always < NNODES
    const bool keep = (depth[row] <= thresh);              // per-lane mask

    const float* __restrict__ xrow = X + (long long)row * DIM + koff;

    v8f acc[TILES_N] = {};
    for (int k = 0; k < DIM; k += 4) {
        v2f a = *(const v2f*)(xrow + k);                   // global_load_b64
        if (do_relu) { a.x = fmaxf(a.x, 0.0f); a.y = fmaxf(a.y, 0.0f); }
        a.x = keep ? a.x : 0.0f;                           // cndmask, EXEC full
        a.y = keep ? a.y : 0.0f;
#pragma unroll
        for (int tn = 0; tn < TILES_N; ++tn) {
            const int col  = tn * 16 + l15;
            const int ccol = (col < DIM) ? col : 0;        // clamp (lanes unused)
            v2f b = *(const v2f*)(&lds_w[ccol * DIM + k + koff]); // ds_load_b64
            acc[tn] = __builtin_amdgcn_wmma_f32_16x16x4_f32(
                          false, a, false, b, (short)0, acc[tn], false, false);
        }
    }

    const int rbase = strip * 16 + ((lane >> 4) << 3);
#pragma unroll
    for (int tn = 0; tn < TILES_N; ++tn) {
        const int col = tn * 16 + l15;
        if (col < DIM) {                                   // divergence after WMMA
            const float bv = bias[col];
#pragma unroll
            for (int r = 0; r < 8; ++r)
                T[(long long)(rbase + r) * DIM + col] = acc[tn][r] + bv;
        }
    }
}

// ---------------------------------------------------------------------------
// Zero the accumulation buffer.
// ---------------------------------------------------------------------------
__global__ __launch_bounds__(256) void gcn_zero(float* __restrict__ p, long long n)
{
    long long i = (long long)blockIdx.x * blockDim.x + threadIdx.x;
    if (i < n) p[i] = 0.0f;
}

// ---------------------------------------------------------------------------
// Edge scatter: agg[d] += T[s] for both directions of every edge.
// One wave per directed edge; lanes cover columns {lane, +32, +64, +96<100}.
// Working set (T + agg + edges ~ 46 MB) is L2-resident (192 MB), so use
// fire-and-forget global_atomic_add_f32 executed at the L2 atomic units.
// ---------------------------------------------------------------------------
__global__ __launch_bounds__(256) void gcn_scatter(
    const float* __restrict__ T,
    const int*   __restrict__ edges,   // [E][2]
    float*       __restrict__ agg)
{
    const int lane = threadIdx.x & 31;
    const long long e = (long long)blockIdx.x * (blockDim.x >> 5) + (threadIdx.x >> 5);
    if (e >= 2LL * NEDGES) return;

    const long long eidx = (e < NEDGES) ? e : (e - NEDGES);
    const int2 uv = ((const int2*)edges)[eidx];            // single b64 load
    const int s = (e < NEDGES) ? uv.x : uv.y;
    const int d = (e < NEDGES) ? uv.y : uv.x;

    const float* __restrict__ src = T   + (long long)s * DIM;
    float*       __restrict__ dst = agg + (long long)d * DIM;

#pragma unroll
    for (int i = 0; i < 4; ++i) {
        const int c = lane + 32 * i;
        if (c < DIM)
            unsafeAtomicAdd(&dst[c], src[c]);  // global_atomic_add_f32, no RET
    }
}

// ---------------------------------------------------------------------------
// Final output mask (layer 3): data = (depth[node] <= 0) ? data : 0, no ReLU.
// (Intermediate masks/ReLUs are folded into the next layer's GEMM A-load.)
// ---------------------------------------------------------------------------
__global__ __launch_bounds__(256) void gcn_final_mask(
    float* __restrict__ data, const int* __restrict__ depth)
{
    long long i = (long long)blockIdx.x * blockDim.x + threadIdx.x;
    if (i >= (long long)NNODES * DIM) return;
    const int node = (int)(i / DIM);
    data[i] = (depth[node] <= 0) ? data[i] : 0.0f;
}

// ---------------------------------------------------------------------------
// 3 layers of (GEMM[+folded prev mask/relu] -> zero -> scatter), final mask.
// d_out doubles as the data/agg buffer; d_ws holds T (20 MB).
// ---------------------------------------------------------------------------
extern "C" void kernel_launch(void* const* d_in, const int* in_sizes, int n_in,
                              void* d_out, int out_size, void* d_ws, size_t ws_size,
                              hipStream_t stream)
{
    const float* x     = (const float*)d_in[0];   // [N,100]
    const float* W     = (const float*)d_in[1];   // [3,100,100]
    const float* bias  = (const float*)d_in[2];   // [3,100]
    const int*   edges = (const int*)d_in[3];     // [E,2]
    const int*   depth = (const int*)d_in[4];     // [N]
    (void)in_sizes; (void)n_in; (void)out_size; (void)ws_size;

    float* data = (float*)d_out;                  // N*DIM f32 (20 MB)
    float* T    = (float*)d_ws;                   // N*DIM f32 (20 MB)

    const int  WPB    = 8;                               // waves per block
    const int  strips = NNODES / 16;                     // 3125
    const int  ggrid  = (strips + WPB - 1) / WPB;        // 391
    const long long nelem = (long long)NNODES * DIM;
    const int  zgrid  = (int)((nelem + 255) / 256);
    const int  sgrid  = (int)((2LL * NEDGES + WPB - 1) / WPB);

    for (int l = 0; l < NLAYERS; ++l) {
        const float* in = (l == 0) ? x : data;
        // Input transform folded from the PREVIOUS layer's mask+relu:
        //   l==0: raw x (thresh = +inf, no relu)
        //   l>=1: keep rows with depth <= NLAYERS - l, relu on.
        const int thresh  = (l == 0) ? 0x7FFFFFFF : (NLAYERS - l);
        const int do_relu = (l == 0) ? 0 : 1;
        gcn_gemm_wmma<<<ggrid, 256, 0, stream>>>(in, W + l * DIM * DIM,
                                                 bias + l * DIM, depth,
                                                 thresh, do_relu, T);
        gcn_zero<<<zgrid, 256, 0, stream>>>(data, nelem);
        gcn_scatter<<<sgrid, 256, 0, stream>>>(T, edges, data);
    }
    gcn_final_mask<<<zgrid, 256, 0, stream>>>(data, depth);
}